// EquiNorm_70875550319297
// MI455X (gfx1250) — compile-verified
//
#include <hip/hip_runtime.h>

typedef float v2f __attribute__((ext_vector_type(2)));
typedef float v8f __attribute__((ext_vector_type(8)));

#define NIMG 64
#define CIN  128
#define COUT 128
#define HW   4096   // 64*64 pixels per image
#define HF   64
#define WF   64

// ---------------------------------------------------------------------------
// Kernel 1: 1x1 conv as GEMM via V_WMMA_F32_16X16X4_F32.
// f[b][p][o] = sum_c x[b][c][p] * w[o][c] + bias[o]   (NHWC output in d_ws)
// One wave: 16 pixels (M) x 128 out-channels (8 N-tiles), K-loop over CIN.
// ---------------------------------------------------------------------------
__global__ __launch_bounds__(256) void conv1x1_wmma(
    const float* __restrict__ x,      // [64][128][4096]
    const float* __restrict__ w,      // [128][128]
    const float* __restrict__ bias,   // [128]
    float* __restrict__ f)            // [64][4096][128]
{
    const int tid  = threadIdx.x;
    const int lane = tid & 31;
    const int wave = blockIdx.x * (blockDim.x >> 5) + (tid >> 5);
    const int b     = wave >> 8;      // image 0..63
    const int pt    = wave & 255;     // pixel tile 0..255
    const int pbase = pt * 16;

    const int mn = lane & 15;         // M index (A) / N index (B) per lane
    const int kb = (lane >> 4) * 2;   // lane-half selects K pair

    const float* xb = x + (size_t)b * CIN * HW;

    v8f acc[8] = {};

#pragma unroll 4
    for (int k0 = 0; k0 < CIN; k0 += 4) {
        // A fragment: A[m][k] = x[b][k][pbase+m]
        v2f a;
        const float* ap = xb + (size_t)(k0 + kb) * HW + pbase + mn;
        a.x = ap[0];      // K = k0+kb
        a.y = ap[HW];     // K = k0+kb+1
#pragma unroll
        for (int j = 0; j < 8; ++j) {
            // B fragment: B[k][n] = w[j*16+n][k]  (contiguous pair -> b64)
            const float* wp = w + (size_t)(j * 16 + mn) * CIN + (k0 + kb);
            v2f bf = *(const v2f*)wp;
            acc[j] = __builtin_amdgcn_wmma_f32_16x16x4_f32(
                false, a, false, bf, (short)0, acc[j], false, false);
        }
    }

    // Epilogue: bias add + NHWC store.  D layout: VGPR r -> M=r (+8 upper half)
    const int mrow = (lane >> 4) * 8;
    float* fb = f + ((size_t)b * HW + pbase) * COUT;
#pragma unroll
    for (int j = 0; j < 8; ++j) {
        float bv = bias[j * 16 + mn];
#pragma unroll
        for (int r = 0; r < 8; ++r) {
            int m = r + mrow;
            fb[(size_t)m * COUT + j * 16 + mn] = acc[j][r] + bv;
        }
    }
}

// ---------------------------------------------------------------------------
// Kernel 2: windowed bilinear splat + weighted average.
// Block = 128 threads (one per channel), 32 target pixels per block.
// LDS transpose (stride 129, conflict-free) for coalesced NCHW stores.
// ---------------------------------------------------------------------------
__global__ __launch_bounds__(128) void splat_avg(
    const float* __restrict__ f,      // [64][64][64][128]  (b, h, w, c)
    const float* __restrict__ win,    // [128][128]
    const float* __restrict__ qs,     // [8][8][2]
    const int*   __restrict__ boxes,  // [8][8][4]
    float* __restrict__ out)          // [8][128][128][128] NCHW
{
    __shared__ float lds[32 * 129];
    const int c  = threadIdx.x;
    const int Xg = blockIdx.x;        // 0..3  (32-pixel group)
    const int Y  = blockIdx.y;        // 0..127
    const int n  = blockIdx.z;        // 0..7

    int bx[8], by[8], bx1[8], by1[8];
    float q1v[8];
#pragma unroll
    for (int k = 0; k < 8; ++k) {
        const int* bp = boxes + (n * 8 + k) * 4;
        bx[k] = bp[0]; by[k] = bp[1]; bx1[k] = bp[2]; by1[k] = bp[3];
        q1v[k] = qs[(n * 8 + k) * 2 + 1];
    }

    for (int px = 0; px < 32; ++px) {
        const int X = Xg * 32 + px;
        float acc = 0.f, wsum = 0.f;
#pragma unroll
        for (int k = 0; k < 8; ++k) {
            int dyi = Y - by[k];
            int dxi = X - bx[k];
            bool inb = (dyi >= 0) && (Y < by1[k]) && (dxi >= 0) && (X < bx1[k]);
            if (!inb) continue;   // weight == 0: contributes nothing

            float h = (float)(by1[k] - by[k]);
            float w = (float)(bx1[k] - bx[k]);
            float denh = fmaxf(h - 1.f, 1.f);
            float denw = fmaxf(w - 1.f, 1.f);
            float dy = (float)dyi, dx = (float)dxi;

            // crop sample coords (Hc=Wc=64)
            float u = dy * 63.f / denh;
            float v = dx * 63.f / denw;
            float u0f = fminf(fmaxf(floorf(u), 0.f), 62.f);
            float v0f = fminf(fmaxf(floorf(v), 0.f), 62.f);
            int   u0  = (int)u0f, v0 = (int)v0f;
            float du  = fminf(fmaxf(u - u0f, 0.f), 1.f);
            float dv  = fminf(fmaxf(v - v0f, 0.f), 1.f);

            float weight;
            if (k == 0) {
                weight = 1.0f;
            } else {
                // window sample coords (Hw=Ww=128)
                float uw = dy * 127.f / denh;
                float vw = dx * 127.f / denw;
                float uw0f = fminf(fmaxf(floorf(uw), 0.f), 126.f);
                float vw0f = fminf(fmaxf(floorf(vw), 0.f), 126.f);
                int   uw0  = (int)uw0f, vw0 = (int)vw0f;
                float duw  = fminf(fmaxf(uw - uw0f, 0.f), 1.f);
                float dvw  = fminf(fmaxf(vw - vw0f, 0.f), 1.f);
                const float* wp = win + uw0 * 128 + vw0;
                float w00 = wp[0], w01 = wp[1], w10 = wp[128], w11 = wp[129];
                weight = (1.f - duw) * ((1.f - dvw) * w00 + dvw * w01)
                       +        duw  * ((1.f - dvw) * w10 + dvw * w11);
            }

            const float* fp = f + ((((size_t)(n * 8 + k) * HF + u0) * WF) + v0) * COUT + c;
            float c00 = fp[0];
            float c01 = fp[COUT];
            float c10 = fp[WF * COUT];
            float c11 = fp[WF * COUT + COUT];
            float val = (1.f - du) * ((1.f - dv) * c00 + dv * c01)
                      +        du  * ((1.f - dv) * c10 + dv * c11);

            acc  += weight * q1v[k] * val;
            wsum += weight;
        }
        lds[px * 129 + c] = acc / fmaxf(wsum, 1e-6f);
    }

    __syncthreads();

    // Transposed write-out: lanes 0..31 write 32 consecutive X for one channel.
#pragma unroll
    for (int i = 0; i < 32; ++i) {
        int idx  = i * 128 + c;
        int c2   = idx >> 5;
        int xoff = idx & 31;
        out[(((size_t)n * COUT + c2) * 128 + Y) * 128 + Xg * 32 + xoff] =
            lds[xoff * 129 + c2];
    }
}

extern "C" void kernel_launch(void* const* d_in, const int* in_sizes, int n_in,
                              void* d_out, int out_size, void* d_ws, size_t ws_size,
                              hipStream_t stream) {
    const float* x      = (const float*)d_in[0];
    const float* conv_w = (const float*)d_in[1];
    const float* conv_b = (const float*)d_in[2];
    const float* win    = (const float*)d_in[3];
    const float* qs     = (const float*)d_in[4];
    const int*   boxes  = (const int*)d_in[5];
    float* f   = (float*)d_ws;   // 64*4096*128 floats = 128 MiB (L2-resident)
    float* out = (float*)d_out;

    // 64 images * 256 pixel-tiles = 16384 waves; 8 waves/block
    conv1x1_wmma<<<2048, 256, 0, stream>>>(x, conv_w, conv_b, f);

    dim3 g2(4, 128, 8);   // Xgroup, Y, n
    splat_avg<<<g2, 128, 0, stream>>>(f, win, qs, boxes, out);
}